// RegionEmbeddingLayer_12421045420642
// MI455X (gfx1250) — compile-verified
//
#include <hip/hip_runtime.h>

// Problem constants (match reference)
#define BB   32
#define SS   2048
#define RGN  7
#define RAD  3
#define DD   128
#define LL   (SS - 2 * RAD)       // 2042
#define TL   32                   // l-positions per block
#define ROWS (TL + RGN - 1)       // 38 word-emb rows staged per block
#define NTHREADS 256

typedef float v4f __attribute__((ext_vector_type(4)));
typedef int   b128_t __attribute__((vector_size(16)));   // matches builtin's V4i pointee

#define AS1 __attribute__((address_space(1)))
#define AS3 __attribute__((address_space(3)))

#if __has_builtin(__builtin_amdgcn_global_load_async_to_lds_b128)
#define HAVE_ASYNC_LDS 1
#else
#define HAVE_ASYNC_LDS 0
#endif

__device__ __forceinline__ void copy16_g2l(const float* gp, float* lp) {
#if HAVE_ASYNC_LDS
    // gfx1250 async DMA: global -> LDS, 16 bytes per active lane, ASYNCcnt-tracked
    __builtin_amdgcn_global_load_async_to_lds_b128(
        (AS1 b128_t*)gp, (AS3 b128_t*)lp, /*offset=*/0, /*cpol=*/0);
#else
    *(v4f*)lp = *(const v4f*)gp;
#endif
}

__device__ __forceinline__ void wait_g2l() {
#if HAVE_ASYNC_LDS
#if __has_builtin(__builtin_amdgcn_s_wait_asynccnt)
    __builtin_amdgcn_s_wait_asynccnt(0);
#else
    asm volatile("s_wait_asynccnt 0" ::: "memory");
#endif
#endif
}

__device__ __forceinline__ v4f max4(v4f a, v4f b) {
    v4f r;
    r.x = fmaxf(a.x, b.x);
    r.y = fmaxf(a.y, b.y);
    r.z = fmaxf(a.z, b.z);
    r.w = fmaxf(a.w, b.w);
    return r;
}

__global__ __launch_bounds__(NTHREADS, 2)
void region_embed_kernel(const int* __restrict__ vids,
                         const float* __restrict__ wemb,
                         const float* __restrict__ cemb,
                         float* __restrict__ out) {
    __shared__ __align__(16) float sW[ROWS * DD];   // 19456 B of the 320 KB WGP LDS

    const int b    = blockIdx.y;
    const int l0   = blockIdx.x * TL;
    const int tid  = threadIdx.x;
    const int lane = tid & 31;
    const int wv   = tid >> 5;          // wave id 0..7
    const int* vrow = vids + b * SS;

    // ---- Stage 38 word-embedding rows (512 B each) into LDS via async DMA ----
    // ROWS * (DD/4) = 1216 16-byte chunks; 256 threads -> <=5 chunks each.
#pragma unroll
    for (int c = tid; c < ROWS * (DD / 4); c += NTHREADS) {
        const int row = c >> 5;                 // 32 chunks per 128-float row
        const int ch4 = (c & 31) * 4;           // float offset within row
        const int s   = l0 + row;
        const int id  = (s < SS) ? vrow[s] : 0; // OOB rows are never consumed
        copy16_g2l(wemb + (size_t)id * DD + ch4, &sW[row * DD + ch4]);
    }
    wait_g2l();
    __syncthreads();

    // ---- Compute: each wave handles one l per iteration, float4 per lane ----
    // Software pipeline: the ctx-row id for iteration k+1 is loaded (and its row
    // prefetched) while iteration k computes, hiding the dependent-gather latency.
    const int lane4 = lane * 4;
    int l   = l0 + wv;
    int idc = (l < LL) ? vrow[l + RAD] : 0;

#pragma unroll
    for (int k = 0; k < TL / 8; ++k) {
        // Kick off next iteration's gather chain early.
        const int ln  = l + 8;
        int idn = 0;
        if (k + 1 < TL / 8 && ln < LL) {
            idn = vrow[ln + RAD];
            if (lane < 28) {   // 28 lanes x 128 B = entire 3584 B ctx row
                __builtin_prefetch((const char*)(cemb + (size_t)idn * (RGN * DD)) +
                                       lane * 128, 0, 0);
            }
        }

        if (l < LL) {
            const int li = l - l0;
            const float* cbase = cemb + (size_t)idc * (RGN * DD) + lane4;
            const float* wbase = &sW[li * DD + lane4];

            v4f acc;
#pragma unroll
            for (int r = 0; r < RGN; ++r) {
                v4f w4 = *(const v4f*)(wbase + r * DD);                            // ds_load_b128
                v4f c4 = __builtin_nontemporal_load((const v4f*)(cbase + r * DD)); // b128 NT load
                v4f p  = w4 * c4;
                acc = (r == 0) ? p : max4(acc, p);                                 // v_max_num_f32
            }
            // Output is write-once: NT store keeps L2 free for the embedding tables.
            __builtin_nontemporal_store(acc,
                (v4f*)(out + ((size_t)b * LL + l) * DD + lane4));                  // b128 NT store
        }

        l   = ln;
        idc = idn;
    }
}

extern "C" void kernel_launch(void* const* d_in, const int* in_sizes, int n_in,
                              void* d_out, int out_size, void* d_ws, size_t ws_size,
                              hipStream_t stream) {
    (void)in_sizes; (void)n_in; (void)d_ws; (void)ws_size; (void)out_size;
    const int*   vids = (const int*)d_in[0];    // vocab_ids (32, 2048)
    const float* wemb = (const float*)d_in[1];  // word_emb (50000, 128)
    const float* cemb = (const float*)d_in[2];  // ctx_emb  (50000, 896)
    float*       out  = (float*)d_out;          // (32, 2042, 128) f32

    dim3 grid((LL + TL - 1) / TL, BB);          // 64 x 32 blocks
    region_embed_kernel<<<grid, NTHREADS, 0, stream>>>(vids, wemb, cemb, out);
}